// AtomicNeuralNetwork_74921409511536
// MI455X (gfx1250) — compile-verified
//
#include <hip/hip_runtime.h>

// Species-routed per-atom MLP for MI455X (gfx1250, wave32, WMMA fp32).
// out[n,a] = MLP_{numbers[a]}(desc[n,a,:])  with layers 39->50->50->1, silu.
//
// Weights are staged in LDS as interleaved K-row pairs so every WMMA B
// fragment is a single aligned ds_load_b64 into a contiguous VGPR pair.

typedef __attribute__((ext_vector_type(2))) float v2f;
typedef __attribute__((ext_vector_type(8))) float v8f;

#define NPTS   4096
#define AATOMS 256
#define DIN    39
#define HDIM   50
#define MB     128          // batch rows per block
#define W1PS   160          // dwords per W1 pair-row (160 mod 64 == 32 -> disjoint banks)
#define W2PS   160          // dwords per W2 pair-row
#define HS     68           // H row stride (even -> aligned b64 A-frags; 68 mod 64 == 4)

__device__ __forceinline__ float silu_f(float x) {
    return x / (1.0f + __expf(-x));
}

__global__ __launch_bounds__(256) void moe_mlp_wmma_kernel(
    const float* __restrict__ desc, const int* __restrict__ numbers,
    const float* __restrict__ W1,   const float* __restrict__ b1,
    const float* __restrict__ W2,   const float* __restrict__ b2,
    const float* __restrict__ W3,   const float* __restrict__ b3,
    float* __restrict__ out)
{
    __shared__ __align__(16) float W1p[20 * W1PS];  // K pairs 0..19 (K=40 pad), N=64 pad
    __shared__ __align__(16) float W2p[26 * W2PS];  // K pairs 0..25 (K=52 pad), N=64 pad
    __shared__ __align__(16) float Hld[MB * HS];    // 128 rows x 64 cols (stride 68)
    __shared__ float w3p[52];
    __shared__ float b1p[64];
    __shared__ float b2p[64];

    const int tid  = threadIdx.x;
    const int lane = tid & 31;
    const int wave = tid >> 5;        // 0..7, each wave owns 16 M rows
    const int m16  = lane & 15;       // A-row index / B,C column index
    const int kh   = lane >> 4;       // K-half selector (lanes 16..31 hold K+2)

    const int chunk = blockIdx.x;     // 0..31  (128-row chunk of N)
    const int atom  = blockIdx.y;     // 0..255
    const int s     = numbers[atom];  // expert id

    // ---- stage expert weights into LDS, K-rows interleaved in pairs ----------
    for (int i = tid; i < 20 * 64; i += 256) {
        const int p = i >> 6, h = i & 63;          // pair p covers K rows 2p, 2p+1
        const int k0r = 2 * p, k1r = 2 * p + 1;
        float x = (k0r < DIN && h < HDIM) ? W1[(s * DIN + k0r) * HDIM + h] : 0.0f;
        float y = (k1r < DIN && h < HDIM) ? W1[(s * DIN + k1r) * HDIM + h] : 0.0f;
        W1p[p * W1PS + 2 * h + 0] = x;
        W1p[p * W1PS + 2 * h + 1] = y;
    }
    for (int i = tid; i < 26 * 64; i += 256) {
        const int p = i >> 6, h = i & 63;
        const int k0r = 2 * p, k1r = 2 * p + 1;
        float x = (k0r < HDIM && h < HDIM) ? W2[(s * HDIM + k0r) * HDIM + h] : 0.0f;
        float y = (k1r < HDIM && h < HDIM) ? W2[(s * HDIM + k1r) * HDIM + h] : 0.0f;
        W2p[p * W2PS + 2 * h + 0] = x;
        W2p[p * W2PS + 2 * h + 1] = y;
    }
    if (tid < 64) {
        b1p[tid] = (tid < HDIM) ? b1[s * HDIM + tid] : 0.0f;
        b2p[tid] = (tid < HDIM) ? b2[s * HDIM + tid] : 0.0f;
    }
    if (tid < 52) w3p[tid] = (tid < HDIM) ? W3[s * HDIM + tid] : 0.0f;
    __syncthreads();

    // ---------------- layer 1: H = silu(desc @ W1 + b1) ------------------------
    const int row0 = chunk * MB + wave * 16;
    const float* descRow = desc + ((size_t)(row0 + m16) * AATOMS + atom) * DIN;

    v8f c0 = {0,0,0,0,0,0,0,0}, c1 = c0, c2 = c0, c3 = c0;
    #pragma unroll
    for (int k0 = 0; k0 < 10; ++k0) {
        const int c = k0 * 4 + kh * 2;                   // this lane's K base
        v2f av;
        av.x = descRow[c];
        av.y = (c + 1 < DIN) ? descRow[c + 1] : 0.0f;    // only k==39 pad
        const v2f* bp = (const v2f*)&W1p[(2 * k0 + kh) * W1PS];
        const v2f bv0 = bp[m16 +  0];
        const v2f bv1 = bp[m16 + 16];
        const v2f bv2 = bp[m16 + 32];
        const v2f bv3 = bp[m16 + 48];
        c0 = __builtin_amdgcn_wmma_f32_16x16x4_f32(false, av, false, bv0, (short)0, c0, false, false);
        c1 = __builtin_amdgcn_wmma_f32_16x16x4_f32(false, av, false, bv1, (short)0, c1, false, false);
        c2 = __builtin_amdgcn_wmma_f32_16x16x4_f32(false, av, false, bv2, (short)0, c2, false, false);
        c3 = __builtin_amdgcn_wmma_f32_16x16x4_f32(false, av, false, bv3, (short)0, c3, false, false);
    }
    {   // bias + silu, write C-layout tiles into LDS H (cols 50..63 become 0)
        float* hbase = &Hld[(wave * 16) * HS];
        #pragma unroll
        for (int r = 0; r < 8; ++r) {
            float* hrow = hbase + (r + 8 * kh) * HS;
            hrow[m16 +  0] = silu_f(c0[r] + b1p[m16 +  0]);
            hrow[m16 + 16] = silu_f(c1[r] + b1p[m16 + 16]);
            hrow[m16 + 32] = silu_f(c2[r] + b1p[m16 + 32]);
            hrow[m16 + 48] = silu_f(c3[r] + b1p[m16 + 48]);
        }
    }

    // ---------------- layer 2: H = silu(H @ W2 + b2) (own slice, in place) ----
    v8f d0 = {0,0,0,0,0,0,0,0}, d1 = d0, d2 = d0, d3 = d0;
    const float* arow = &Hld[(wave * 16 + m16) * HS];
    #pragma unroll
    for (int k0 = 0; k0 < 13; ++k0) {
        const int c = k0 * 4 + kh * 2;                   // <= 50; cols 50,51 are zeros
        const v2f av = *(const v2f*)(arow + c);          // aligned b64 (HS even, c even)
        const v2f* bp = (const v2f*)&W2p[(2 * k0 + kh) * W2PS];
        const v2f bv0 = bp[m16 +  0];
        const v2f bv1 = bp[m16 + 16];
        const v2f bv2 = bp[m16 + 32];
        const v2f bv3 = bp[m16 + 48];
        d0 = __builtin_amdgcn_wmma_f32_16x16x4_f32(false, av, false, bv0, (short)0, d0, false, false);
        d1 = __builtin_amdgcn_wmma_f32_16x16x4_f32(false, av, false, bv1, (short)0, d1, false, false);
        d2 = __builtin_amdgcn_wmma_f32_16x16x4_f32(false, av, false, bv2, (short)0, d2, false, false);
        d3 = __builtin_amdgcn_wmma_f32_16x16x4_f32(false, av, false, bv3, (short)0, d3, false, false);
    }
    {   // bias + silu, overwrite own H slice (DS ops are in-order per wave)
        float* hbase = &Hld[(wave * 16) * HS];
        #pragma unroll
        for (int r = 0; r < 8; ++r) {
            float* hrow = hbase + (r + 8 * kh) * HS;
            hrow[m16 +  0] = silu_f(d0[r] + b2p[m16 +  0]);
            hrow[m16 + 16] = silu_f(d1[r] + b2p[m16 + 16]);
            hrow[m16 + 32] = silu_f(d2[r] + b2p[m16 + 32]);
            hrow[m16 + 48] = silu_f(d3[r] + b2p[m16 + 48]);
        }
    }
    __syncthreads();

    // ---------------- layer 3: out = H @ w3 + b3 (M x 1, VALU dot) -------------
    if (tid < MB) {
        float acc = b3[s];
        const float* hrow = &Hld[tid * HS];
        #pragma unroll
        for (int h = 0; h < HDIM; ++h) acc += hrow[h] * w3p[h];
        out[(size_t)(chunk * MB + tid) * AATOMS + atom] = acc;
    }
}

extern "C" void kernel_launch(void* const* d_in, const int* in_sizes, int n_in,
                              void* d_out, int out_size, void* d_ws, size_t ws_size,
                              hipStream_t stream) {
    (void)in_sizes; (void)n_in; (void)d_ws; (void)ws_size; (void)out_size;
    const float* desc    = (const float*)d_in[0];
    const int*   numbers = (const int*)  d_in[1];
    const float* W1      = (const float*)d_in[2];
    const float* b1      = (const float*)d_in[3];
    const float* W2      = (const float*)d_in[4];
    const float* b2      = (const float*)d_in[5];
    const float* W3      = (const float*)d_in[6];
    const float* b3      = (const float*)d_in[7];
    float* out = (float*)d_out;

    dim3 grid(NPTS / MB, AATOMS);   // (32, 256)
    moe_mlp_wmma_kernel<<<grid, 256, 0, stream>>>(desc, numbers, W1, b1, W2, b2, W3, b3, out);
}